// GNN_22900765622531
// MI455X (gfx1250) — compile-verified
//
#include <hip/hip_runtime.h>

// Problem constants (match reference)
constexpr int N  = 100000;      // nodes
constexpr int E  = 1600000;     // edges (without self loops)
constexpr int ET = E + N;       // edges + self loops
constexpr int H  = 128;         // hidden
constexpr int A  = 64;          // actions
constexpr float NEG_SLOPE = 0.2f;
constexpr float LN_EPS = 1e-5f;
constexpr float SM_EPS = 1e-16f;

typedef __attribute__((ext_vector_type(2))) float v2f;
typedef __attribute__((ext_vector_type(8))) float v8f;

__device__ __forceinline__ void atomicMaxF(float* addr, float v) {
    // standard monotonic-bit-pattern trick; memory initialized to -inf
    if (v >= 0.0f) atomicMax((int*)addr, __float_as_int(v));
    else           atomicMin((unsigned int*)addr, (unsigned int)__float_as_int(v));
}

__device__ __forceinline__ float leaky(float v) {
    return v >= 0.0f ? v : NEG_SLOPE * v;
}

__device__ __forceinline__ void edge_sd(const int* ei, int e, int& s, int& d) {
    if (e < E) { s = ei[e]; d = ei[E + e]; }
    else       { s = d = e - E; }
}

// ---------------- init: clear accumulators, set maxes to -inf, zero d_out ----
__global__ void k_init(float* m1, float* den1, float* num1,
                       float* m2, float* den2, float* num2, float* out) {
    long i = (long)blockIdx.x * blockDim.x + threadIdx.x;
    if (i < (long)N * A) num2[i] = 0.0f;
    if (i < N) {
        float ninf = __int_as_float(0xff800000);
        m1[i] = ninf; m2[i] = ninf;
        den1[i] = 0.0f; den2[i] = 0.0f; num1[i] = 0.0f;
    }
    if (i < A) out[i] = 0.0f;
}

// ---------------- c_s = w1·att_src1 ; c_d = w1·att_dst1 ---------------------
__global__ void k_prep(const float* W1, const float* as1, const float* ad1, float* c) {
    __shared__ float red[2 * H];
    int t = threadIdx.x;                 // 128 threads
    float w = W1[t];
    red[t]     = w * as1[t];
    red[H + t] = w * ad1[t];
    __syncthreads();
    for (int s = H / 2; s > 0; s >>= 1) {
        if (t < s) { red[t] += red[t + s]; red[H + t] += red[H + t + s]; }
        __syncthreads();
    }
    if (t == 0) { c[0] = red[0]; c[1] = red[H]; }
}

// ---------------- layer-1 edge passes (scalar, thanks to rank-1 W1) ---------
__global__ void k_edge1_max(const float* x, const int* ei, const float* c, float* m1) {
    int e = blockIdx.x * blockDim.x + threadIdx.x;
    if (e >= ET) return;
    int s, d; edge_sd(ei, e, s, d);
    atomicMaxF(&m1[d], leaky(x[s] * c[0] + x[d] * c[1]));
}

__global__ void k_edge1_sum(const float* x, const int* ei, const float* c,
                            const float* m1, float* den1, float* num1) {
    int e = blockIdx.x * blockDim.x + threadIdx.x;
    if (e >= ET) return;
    int s, d; edge_sd(ei, e, s, d);
    float xs = x[s];
    float ex = expf(leaky(xs * c[0] + x[d] * c[1]) - m1[d]);
    atomicAdd(&den1[d], ex);
    atomicAdd(&num1[d], ex * xs);
}

__global__ void k_node1(const float* num1, const float* den1, float* s1) {
    int i = blockIdx.x * blockDim.x + threadIdx.x;
    if (i < N) s1[i] = num1[i] / (den1[i] + SM_EPS);
}

// ---------------- layer-2 linear: h2 = relu(s1⊗w1 + b1) @ W2 via f32 WMMA ---
// Block: 256 threads = 8 waves; tile 32 rows x 64 cols; each wave = 16x16 tile.
__global__ void __launch_bounds__(256) k_gemm(const float* s1, const float* W1,
                                              const float* b1, const float* W2,
                                              float* h2) {
    __shared__ float sh_h[32 * 132];   // padded stride 132 (bank-conflict free)
    __shared__ float sh_w[H * 65];     // padded stride 65
    const int t = threadIdx.x;
    const int base = blockIdx.x * 32;

    for (int i = t; i < H * A; i += 256) {
        int r = i >> 6, cc = i & 63;
        sh_w[r * 65 + cc] = W2[i];
    }
    for (int i = t; i < 32 * H; i += 256) {
        int r = i >> 7, k = i & 127;
        int node = base + r;
        float v = 0.0f;
        if (node < N) {
            v = s1[node] * W1[k] + b1[k];
            v = v > 0.0f ? v : 0.0f;           // relu (layer-1 activation)
        }
        sh_h[r * 132 + k] = v;
    }
    __syncthreads();

    const int wave = t >> 5;
    const int lane = t & 31;
    const int rowt = (wave >> 2) * 16;     // 0 or 16
    const int colt = (wave & 3) * 16;      // 0,16,32,48
    const int mn = lane & 15;              // row of A / col of B
    const int hi = lane >> 4;              // K sub-pair select

    v8f c = {};
    #pragma unroll
    for (int k0 = 0; k0 < H; k0 += 4) {
        const int ka = k0 + hi * 2;
        v2f a, b;
        a.x = sh_h[(rowt + mn) * 132 + ka];
        a.y = sh_h[(rowt + mn) * 132 + ka + 1];
        b.x = sh_w[ka * 65 + colt + mn];
        b.y = sh_w[(ka + 1) * 65 + colt + mn];
        c = __builtin_amdgcn_wmma_f32_16x16x4_f32(false, a, false, b,
                                                  (short)0, c, false, false);
    }

    const int col = colt + mn;
    const int rbase = rowt + hi * 8;
    #pragma unroll
    for (int vi = 0; vi < 8; vi++) {
        int node = base + rbase + vi;
        if (node < N) h2[(long)node * A + col] = c[vi];
    }
}

// ---------------- per-node attention dots for layer 2 -----------------------
__global__ void k_att2(const float* h2, const float* as2, const float* ad2,
                       float* asrc, float* adst) {
    int i = blockIdx.x * blockDim.x + threadIdx.x;
    if (i >= N) return;
    const float* h = h2 + (long)i * A;
    float s = 0.0f, d = 0.0f;
    #pragma unroll 8
    for (int k = 0; k < A; k++) { float v = h[k]; s += v * as2[k]; d += v * ad2[k]; }
    asrc[i] = s; adst[i] = d;
}

// ---------------- layer-2 edge passes ---------------------------------------
__global__ void k_edge2_max(const int* ei, const float* asrc, const float* adst,
                            float* m2) {
    int e = blockIdx.x * blockDim.x + threadIdx.x;
    if (e >= ET) return;
    int s, d; edge_sd(ei, e, s, d);
    atomicMaxF(&m2[d], leaky(asrc[s] + adst[d]));
}

__global__ void k_edge2_exp(const int* ei, const float* asrc, const float* adst,
                            const float* m2, float* den2, float* exbuf) {
    int e = blockIdx.x * blockDim.x + threadIdx.x;
    if (e >= ET) return;
    int s, d; edge_sd(ei, e, s, d);
    float ex = expf(leaky(asrc[s] + adst[d]) - m2[d]);
    exbuf[e] = ex;
    atomicAdd(&den2[d], ex);
}

__global__ void k_edge2_agg(const int* ei, const float* exbuf, const float* h2,
                            float* num2) {
    long i = (long)blockIdx.x * blockDim.x + threadIdx.x;
    if (i >= (long)ET * A) return;
    int e = (int)(i >> 6), k = (int)(i & 63);
    int s, d; edge_sd(ei, e, s, d);
    atomicAdd(&num2[(long)d * A + k], exbuf[e] * h2[(long)s * A + k]);
}

// ---------------- finalize: relu -> LayerNorm -> global sum -----------------
__global__ void __launch_bounds__(64) k_final(const float* num2, const float* den2,
                                              const float* b2, const float* gamma,
                                              const float* beta, float* out) {
    __shared__ float p[4];
    const int t = threadIdx.x;             // feature index, 64 threads = 2 waves
    const float g = gamma[t], be = beta[t], bb = b2[t];
    float acc = 0.0f;
    for (int node = blockIdx.x; node < N; node += gridDim.x) {
        float v = num2[(long)node * A + t] / (den2[node] + SM_EPS) + bb;
        v = v > 0.0f ? v : 0.0f;
        float sum = v, sq = v * v;
        #pragma unroll
        for (int o = 16; o > 0; o >>= 1) {
            sum += __shfl_xor(sum, o, 32);
            sq  += __shfl_xor(sq,  o, 32);
        }
        if ((t & 31) == 0) { p[t >> 5] = sum; p[2 + (t >> 5)] = sq; }
        __syncthreads();
        float tsum = p[0] + p[1], tsq = p[2] + p[3];
        __syncthreads();
        float mu  = tsum * (1.0f / A);
        float var = tsq * (1.0f / A) - mu * mu;
        acc += (v - mu) * rsqrtf(var + LN_EPS) * g + be;
    }
    atomicAdd(&out[t], acc);
}

extern "C" void kernel_launch(void* const* d_in, const int* in_sizes, int n_in,
                              void* d_out, int out_size, void* d_ws, size_t ws_size,
                              hipStream_t stream) {
    const float* x    = (const float*)d_in[0];
    const int*   ei   = (const int*)  d_in[1];
    const float* W1   = (const float*)d_in[2];
    const float* as1  = (const float*)d_in[3];
    const float* ad1  = (const float*)d_in[4];
    const float* b1   = (const float*)d_in[5];
    const float* W2   = (const float*)d_in[6];
    const float* as2  = (const float*)d_in[7];
    const float* ad2  = (const float*)d_in[8];
    const float* b2   = (const float*)d_in[9];
    const float* gam  = (const float*)d_in[10];
    const float* bet  = (const float*)d_in[11];
    float* out = (float*)d_out;

    float* w = (float*)d_ws;
    float* c     = w;                     // 2 scalars (pad 16)
    float* m1    = w + 16;                // [N]
    float* den1  = m1   + N;              // [N]
    float* num1  = den1 + N;              // [N]
    float* s1    = num1 + N;              // [N]
    float* asrc  = s1   + N;              // [N]
    float* adst  = asrc + N;              // [N]
    float* m2    = adst + N;              // [N]
    float* den2  = m2   + N;              // [N]
    float* h2    = den2 + N;              // [N*64]
    float* num2  = h2   + (long)N * A;    // [N*64]
    float* exbuf = num2 + (long)N * A;    // [ET]

    const int TB = 256;
    k_init<<<(int)(((long)N * A + TB - 1) / TB), TB, 0, stream>>>(
        m1, den1, num1, m2, den2, num2, out);
    k_prep<<<1, H, 0, stream>>>(W1, as1, ad1, c);
    k_edge1_max<<<(ET + TB - 1) / TB, TB, 0, stream>>>(x, ei, c, m1);
    k_edge1_sum<<<(ET + TB - 1) / TB, TB, 0, stream>>>(x, ei, c, m1, den1, num1);
    k_node1<<<(N + TB - 1) / TB, TB, 0, stream>>>(num1, den1, s1);
    k_gemm<<<(N + 31) / 32, 256, 0, stream>>>(s1, W1, b1, W2, h2);
    k_att2<<<(N + TB - 1) / TB, TB, 0, stream>>>(h2, as2, ad2, asrc, adst);
    k_edge2_max<<<(ET + TB - 1) / TB, TB, 0, stream>>>(ei, asrc, adst, m2);
    k_edge2_exp<<<(ET + TB - 1) / TB, TB, 0, stream>>>(ei, asrc, adst, m2, den2, exbuf);
    k_edge2_agg<<<(int)(((long)ET * A + TB - 1) / TB), TB, 0, stream>>>(
        ei, exbuf, h2, num2);
    k_final<<<1024, 64, 0, stream>>>(num2, den2, b2, gam, bet, out);
}